// WaveletTransform2D_Level1_85779086836065
// MI455X (gfx1250) — compile-verified
//
#include <hip/hip_runtime.h>

typedef float v4f __attribute__((ext_vector_type(4)));

// Problem geometry (from reference): x is (8,64,512,512) f32.
#define PLANES        512          // 8*64 collapsed
#define W_IN          512
#define IN_PLANE      (512*512)    // 262144 floats
#define W_OUT         256
#define OUT_PLANE     (256*256)    // 65536 floats
#define OUT_TENSOR    (PLANES*OUT_PLANE)   // 33554432 floats per sub-band

#define BLOCK         256
#define ROWS_PER_ITER 4            // output rows per iteration (8 input rows, contiguous)
#define NITER         4            // iterations per block
#define ROWS_PER_BLK  (ROWS_PER_ITER*NITER)     // 16 output rows per block
#define STRIP_FLOATS  (ROWS_PER_ITER*2*W_IN)    // 4096 floats = 16 KB
#define STRIP_BYTES   (STRIP_FLOATS*4)

// Issue one 16KB strip (global -> LDS) with 4 async b128 loads per thread.
// GVS mode: 64-bit SGPR base + 32-bit per-lane VGPR byte offset.
// Input is read exactly once -> non-temporal load hint (don't rinse L2).
__device__ __forceinline__ void issue_strip_async(const float* strip_base,
                                                  unsigned lds_byte_base,
                                                  unsigned tid)
{
    unsigned long long sb = (unsigned long long)(uintptr_t)strip_base;
#pragma unroll
    for (int k = 0; k < 4; ++k) {
        unsigned goff = (unsigned)k * 4096u + tid * 16u;   // byte offset in strip
        unsigned loff = lds_byte_base + goff;              // LDS byte address
        asm volatile("global_load_async_to_lds_b128 %0, %1, %2 th:TH_LOAD_NT"
                     :
                     : "v"(loff), "v"(goff), "s"(sb)
                     : "memory");
    }
}

__global__ __launch_bounds__(BLOCK)
void haar2d_lvl1_kernel(const float* __restrict__ x, float* __restrict__ out)
{
    __shared__ float smem[2 * STRIP_FLOATS];   // 32 KB double buffer

    const unsigned tid   = threadIdx.x;
    const unsigned plane = blockIdx.x >> 4;          // 16 row-blocks per plane
    const unsigned rb    = blockIdx.x & 15u;

    // Float offset of this block's first strip inside the input tensor.
    const unsigned in_base = plane * (unsigned)IN_PLANE + rb * (ROWS_PER_BLK * 2 * W_IN);

    // LDS byte address of smem[0] (low 32 bits of the flat pointer == LDS offset).
    const unsigned lds0 = (unsigned)(uintptr_t)&smem[0];

    // Prologue: prefetch strip 0 into buffer 0.
    issue_strip_async(x + in_base, lds0, tid);

    const unsigned g = tid >> 6;     // output row within iteration (0..3)
    const unsigned u = tid & 63u;    // output col group (4 cols each)

    float* const o_ll = out;
    float* const o_hl = out + (size_t)OUT_TENSOR;
    float* const o_lh = out + 2 * (size_t)OUT_TENSOR;
    float* const o_hh = out + 3 * (size_t)OUT_TENSOR;

#pragma unroll
    for (int j = 0; j < NITER; ++j) {
        const unsigned buf = (unsigned)j & 1u;

        if (j + 1 < NITER) {
            // Prefetch next strip into the other buffer, then wait for the
            // current strip (async loads complete in order: <=4 outstanding
            // means the 4 loads of strip j have landed in LDS).
            issue_strip_async(x + in_base + (unsigned)(j + 1) * STRIP_FLOATS,
                              lds0 + (buf ^ 1u) * STRIP_BYTES, tid);
            asm volatile("s_wait_asynccnt 0x4" ::: "memory");
        } else {
            asm volatile("s_wait_asynccnt 0x0" ::: "memory");
        }
        __syncthreads();   // all waves' strip-j data visible in LDS

        // Two adjacent input rows for output row (rb*16 + j*4 + g).
        const unsigned sb0 = buf * STRIP_FLOATS + g * (2 * W_IN) + u * 8u;
        const v4f r0a = *(const v4f*)&smem[sb0 + 0];
        const v4f r0b = *(const v4f*)&smem[sb0 + 4];
        const v4f r1a = *(const v4f*)&smem[sb0 + W_IN + 0];
        const v4f r1b = *(const v4f*)&smem[sb0 + W_IN + 4];

        float r0[8], r1[8];
        *(v4f*)&r0[0] = r0a; *(v4f*)&r0[4] = r0b;
        *(v4f*)&r1[0] = r1a; *(v4f*)&r1[4] = r1b;

        v4f ll, hl, lh, hh;
#pragma unroll
        for (int k = 0; k < 4; ++k) {
            const float a = r0[2 * k], b = r0[2 * k + 1];
            const float c = r1[2 * k], d = r1[2 * k + 1];
            const float s = a + b, t = c + d;   // row sums
            const float p = a - b, q = c - d;   // row diffs
            ll[k] = (s + t) * 0.5f;
            hl[k] = (s - t) * 0.5f;
            lh[k] = (p + q) * 0.5f;
            hh[k] = (p - q) * 0.5f;
        }

        const unsigned row  = rb * ROWS_PER_BLK + (unsigned)j * ROWS_PER_ITER + g;
        const size_t   ooff = (size_t)plane * OUT_PLANE + (size_t)row * W_OUT + u * 4u;

        // Streaming outputs: never re-read -> non-temporal stores.
        __builtin_nontemporal_store(ll, (v4f*)(o_ll + ooff));
        __builtin_nontemporal_store(hl, (v4f*)(o_hl + ooff));
        __builtin_nontemporal_store(lh, (v4f*)(o_lh + ooff));
        __builtin_nontemporal_store(hh, (v4f*)(o_hh + ooff));

        if (j + 1 < NITER)
            __syncthreads();   // strip-j buffer free for reuse two iters later
    }
}

extern "C" void kernel_launch(void* const* d_in, const int* in_sizes, int n_in,
                              void* d_out, int out_size, void* d_ws, size_t ws_size,
                              hipStream_t stream)
{
    (void)in_sizes; (void)n_in; (void)d_ws; (void)ws_size; (void)out_size;
    const float* x  = (const float*)d_in[0];
    float*       out = (float*)d_out;

    dim3 grid(PLANES * (256 / ROWS_PER_BLK));   // 512 * 16 = 8192 blocks
    dim3 block(BLOCK);
    haar2d_lvl1_kernel<<<grid, block, 0, stream>>>(x, out);
}